// GCNDecoder_45500883534472
// MI455X (gfx1250) — compile-verified
//
#include <hip/hip_runtime.h>

// ---------------------------------------------------------------------------
// GCNConv: out = scatter_add( norm * (z@W)[src] -> dst ) + selfloop + bias
// N=100000 nodes, E=1600000 edges, D_IN=D_OUT=64, fp32 throughout.
// GEMM runs on the CDNA5 matrix pipe via V_WMMA_F32_16X16X4_F32 (wave32).
// W is staged in LDS with a K-pair-interleaved layout so each B fragment is
// one aligned ds_load_b64 (no cross-register packing moves in the hot loop).
// ---------------------------------------------------------------------------

typedef float v2f __attribute__((ext_vector_type(2)));
typedef float v8f __attribute__((ext_vector_type(8)));

#define D 64            // feature dim (in == out == 64)
#define TILE_M 16       // rows per wave
#define WAVES_PER_BLOCK 8

// ---- 1) zero the output, init degree with self-loop (deg = 1) -------------
__global__ void gcn_init_kernel(float4* __restrict__ out4, float* __restrict__ deg,
                                int n_nodes) {
    int t = blockIdx.x * blockDim.x + threadIdx.x;
    int n_out4 = n_nodes * (D / 4);
    if (t < n_out4) out4[t] = make_float4(0.f, 0.f, 0.f, 0.f);
    if (t < n_nodes) deg[t] = 1.0f;      // self-loop contribution
}

// ---- 2) degree at destination: deg[dst] += 1 -------------------------------
__global__ void gcn_degree_kernel(const int* __restrict__ dst, float* __restrict__ deg,
                                  int n_edges) {
    int e = blockIdx.x * blockDim.x + threadIdx.x;
    if (e >= n_edges) return;
    unsafeAtomicAdd(&deg[dst[e]], 1.0f);   // global_atomic_add_f32
}

// ---- 3) deg -> deg^{-1/2} in place (deg >= 1 always, self-loops) ----------
__global__ void gcn_rsqrt_kernel(float* __restrict__ deg, int n_nodes) {
    int i = blockIdx.x * blockDim.x + threadIdx.x;
    if (i >= n_nodes) return;
    deg[i] = __frsqrt_rn(deg[i]);
}

// ---- 4) x = z @ W via V_WMMA_F32_16X16X4_F32 ------------------------------
// One wave computes a 16x64 slab: 4 accumulators (16x16), K=64 in 16 steps of 4.
// A frag (16x4 f32): lane<16 -> rows M=lane, K pair (k,k+1); lane>=16 -> (k+2,k+3).
//   -> single global_load_b64 per step (zrow + k, k even).
// B frag (4x16 f32): v.x = W[k][n], v.y = W[k+1][n], n = nb*16 + lane%16,
//                    k = 4*kb + 2*(lane/16).
//   -> LDS holds W interleaved by K-pair: Wlds[((k>>1)*64 + n)*2 + (k&1)],
//      so the fragment is one aligned ds_load_b64 at (kp*64 + n)*8 bytes.
// D (16x16 f32): VGPR r -> row m0 + r + 8*(lane/16), col lane%16.
__global__ void gcn_gemm_wmma_kernel(const float* __restrict__ z,
                                     const float* __restrict__ w,
                                     float* __restrict__ x, int n_nodes) {
    __shared__ float Wlds[D * D];   // 16 KB, K-pair-interleaved layout
    for (int i = threadIdx.x; i < D * D; i += blockDim.x) {
        int k = i >> 6;             // row of W
        int n = i & 63;             // col of W
        Wlds[((((k >> 1) << 6) + n) << 1) + (k & 1)] = w[i];
    }
    __syncthreads();

    const int lane  = threadIdx.x & 31;
    const int wave  = threadIdx.x >> 5;
    const int tile  = blockIdx.x * WAVES_PER_BLOCK + wave;
    const int m0    = tile * TILE_M;
    if (m0 >= n_nodes) return;

    const int mrow  = lane & 15;        // row within tile / col within N-tile
    const int khalf = (lane >> 4) << 1; // 0 or 2

    v8f acc0 = {}, acc1 = {}, acc2 = {}, acc3 = {};
    const float* __restrict__ zrow = z + (size_t)(m0 + mrow) * D;
    const v2f* __restrict__ Wp = (const v2f*)Wlds;   // [kp*64 + n] -> (W[k][n], W[k+1][n])

    #pragma unroll
    for (int kb = 0; kb < D / 4; ++kb) {
        const int k  = (kb << 2) + khalf;
        const int kp = (kb << 1) + (lane >> 4);      // K-pair index = k>>1

        const v2f a = *(const v2f*)(zrow + k);       // global_load_b64
        const v2f* __restrict__ bp = Wp + (kp << 6) + mrow;

        acc0 = __builtin_amdgcn_wmma_f32_16x16x4_f32(false, a, false, bp[0],
                                                     (short)0, acc0, false, false);
        acc1 = __builtin_amdgcn_wmma_f32_16x16x4_f32(false, a, false, bp[16],
                                                     (short)0, acc1, false, false);
        acc2 = __builtin_amdgcn_wmma_f32_16x16x4_f32(false, a, false, bp[32],
                                                     (short)0, acc2, false, false);
        acc3 = __builtin_amdgcn_wmma_f32_16x16x4_f32(false, a, false, bp[48],
                                                     (short)0, acc3, false, false);
    }

    // store: VGPR r of acc{nb} -> x[m0 + r + 8*(lane>>4)][nb*16 + mrow]
    const int rbase = m0 + ((lane >> 4) << 3);
    #pragma unroll
    for (int r = 0; r < 8; ++r) {
        float* __restrict__ orow = x + (size_t)(rbase + r) * D + mrow;
        orow[ 0] = acc0[r];
        orow[16] = acc1[r];
        orow[32] = acc2[r];
        orow[48] = acc3[r];
    }
}

// ---- 5) edge scatter: out[dst] += dis[src]*dis[dst] * x[src] --------------
// 16 lanes per edge, each lane owns a float4 of the 64-wide feature row.
__global__ void gcn_scatter_kernel(const int* __restrict__ src,
                                   const int* __restrict__ dst,
                                   const float* __restrict__ x,
                                   const float* __restrict__ dis,
                                   float* __restrict__ out, int n_edges) {
    int t = blockIdx.x * blockDim.x + threadIdx.x;
    int e = t >> 4;
    if (e >= n_edges) return;
    int c4 = (t & 15) << 2;

    int s = src[e];
    int d = dst[e];
    float norm = dis[s] * dis[d];

    const float4 xv = *(const float4*)(x + (size_t)s * D + c4);
    float* op = out + (size_t)d * D + c4;
    unsafeAtomicAdd(op + 0, xv.x * norm);
    unsafeAtomicAdd(op + 1, xv.y * norm);
    unsafeAtomicAdd(op + 2, xv.z * norm);
    unsafeAtomicAdd(op + 3, xv.w * norm);
}

// ---- 6) self-loop (norm = dis[i]^2) + bias --------------------------------
__global__ void gcn_finalize_kernel(float* __restrict__ out,
                                    const float* __restrict__ x,
                                    const float* __restrict__ dis,
                                    const float* __restrict__ bias, int n_nodes) {
    int t = blockIdx.x * blockDim.x + threadIdx.x;
    if (t >= n_nodes * (D / 4)) return;
    int i  = t >> 4;
    int c4 = (t & 15) << 2;

    float self_norm = dis[i] * dis[i];
    float4 o = *(float4*)(out + (size_t)i * D + c4);
    float4 xv = *(const float4*)(x + (size_t)i * D + c4);
    float4 b  = *(const float4*)(bias + c4);
    o.x += self_norm * xv.x + b.x;
    o.y += self_norm * xv.y + b.y;
    o.z += self_norm * xv.z + b.z;
    o.w += self_norm * xv.w + b.w;
    *(float4*)(out + (size_t)i * D + c4) = o;
}

// ---------------------------------------------------------------------------
extern "C" void kernel_launch(void* const* d_in, const int* in_sizes, int n_in,
                              void* d_out, int out_size, void* d_ws, size_t ws_size,
                              hipStream_t stream) {
    const float* z     = (const float*)d_in[0];              // [N, 64]
    const int*   edges = (const int*)d_in[1];                // [2, E] row-major
    const float* w     = (const float*)d_in[2];              // [64, 64]
    const float* bias  = (const float*)d_in[3];              // [64]
    float*       out   = (float*)d_out;                      // [N, 64]

    const int N = in_sizes[0] / D;
    const int E = in_sizes[1] / 2;
    const int* e_src = edges;
    const int* e_dst = edges + E;

    // workspace layout: x [N*64 f32] | deg/dis [N f32]
    float* x   = (float*)d_ws;
    float* deg = x + (size_t)N * D;

    const int B = 256;

    // 1) zero out, deg = 1 (self loop)
    {
        int nt = N * (D / 4);
        gcn_init_kernel<<<(nt + B - 1) / B, B, 0, stream>>>((float4*)out, deg, N);
    }
    // 2) degree at dst
    gcn_degree_kernel<<<(E + B - 1) / B, B, 0, stream>>>(e_dst, deg, E);
    // 3) deg -> deg^{-1/2}
    gcn_rsqrt_kernel<<<(N + B - 1) / B, B, 0, stream>>>(deg, N);
    // 4) x = z @ W  (WMMA)
    {
        int tiles  = (N + TILE_M - 1) / TILE_M;
        int blocks = (tiles + WAVES_PER_BLOCK - 1) / WAVES_PER_BLOCK;
        gcn_gemm_wmma_kernel<<<blocks, WAVES_PER_BLOCK * 32, 0, stream>>>(z, w, x, N);
    }
    // 5) edge scatter (hardware f32 atomics, L2-resident)
    {
        long long nt = (long long)E * 16;
        gcn_scatter_kernel<<<(int)((nt + B - 1) / B), B, 0, stream>>>(e_src, e_dst, x,
                                                                      deg, out, E);
    }
    // 6) self-loop + bias
    {
        int nt = N * (D / 4);
        gcn_finalize_kernel<<<(nt + B - 1) / B, B, 0, stream>>>(out, x, deg, bias, N);
    }
}